// Communication_64467459113042
// MI455X (gfx1250) — compile-verified
//
#include <hip/hip_runtime.h>

typedef __attribute__((ext_vector_type(2))) float v2f;
typedef __attribute__((ext_vector_type(8))) float v8f;

#define MAXN 100
#define NCH  256
#define HH   256
#define WW   256
#define HW   (HH * WW)

// ---------------------------------------------------------------------------
// Stage 1: per-box stats + grid-sample + rank-4 coefficient reduction.
// One block, 256 threads. Output: P[256][4] f32 in d_ws.
// ---------------------------------------------------------------------------
__global__ void __launch_bounds__(256)
comm_stage1(const float* __restrict__ box,    // [100][8][3]
            const float* __restrict__ score,  // [100]
            const float* __restrict__ feat,   // [256][256][256]
            float* __restrict__ P)            // [256][4]
{
    __shared__ int   s_idx[MAXN];
    __shared__ int   s_off[MAXN][4];
    __shared__ float s_wgt[MAXN][4];
    __shared__ float s_coef[MAXN][4];

    const int t = threadIdx.x;

    // nonzero(score > 0, size=100, fill_value=0): ordered scan, serial (tiny).
    if (t == 0) {
        int cnt = 0;
        for (int i = 0; i < MAXN; ++i)
            if (score[i] > 0.0f) s_idx[cnt++] = i;
        for (; cnt < MAXN; ++cnt) s_idx[cnt] = 0;
    }
    __syncthreads();

    if (t < MAXN) {
        const float* bp = box + s_idx[t] * 24;
        float lx = bp[0], ly = bp[1], rx = bp[0], ry = bp[1];
        #pragma unroll
        for (int k = 1; k < 8; ++k) {
            float x = bp[k * 3 + 0], y = bp[k * 3 + 1];
            lx = fminf(lx, x); rx = fmaxf(rx, x);
            ly = fminf(ly, y); ry = fmaxf(ry, y);
        }
        const float cx  = ((lx + rx) * 0.5f + 128.0f) / 160.0f;
        const float cy  = ((ly + ry) * 0.5f + 128.0f) / 160.0f;
        const float lxn = (lx + 128.0f) / 160.0f, lyn = (ly + 128.0f) / 160.0f;
        const float rxn = (rx + 128.0f) / 160.0f, ryn = (ry + 128.0f) / 160.0f;
        const float bev = (ryn - lyn) * (rxn - lxn);
        const float s   = 1.0f / (2.0f * bev * bev);

        // grid_sample (bilinear, zero pad, align_corners=False)
        const float ix = ((cx + 1.0f) * (float)WW - 1.0f) * 0.5f;
        const float iy = ((cy + 1.0f) * (float)HH - 1.0f) * 0.5f;
        const float x0f = floorf(ix), y0f = floorf(iy);
        const int   x0 = (int)x0f, y0 = (int)y0f;
        const float wx1 = ix - x0f, wx0 = 1.0f - wx1;
        const float wy1 = iy - y0f, wy0 = 1.0f - wy1;
        const int   xs[2]  = { x0, x0 + 1 };
        const int   ys[2]  = { y0, y0 + 1 };
        const float wxs[2] = { wx0, wx1 };
        const float wys[2] = { wy0, wy1 };
        #pragma unroll
        for (int jy = 0; jy < 2; ++jy) {
            #pragma unroll
            for (int jx = 0; jx < 2; ++jx) {
                const int  j  = jy * 2 + jx;
                const int  xi = xs[jx], yi = ys[jy];
                const bool ok = (xi >= 0) && (xi < WW) && (yi >= 0) && (yi < HH);
                const int  xc = min(max(xi, 0), WW - 1);
                const int  yc = min(max(yi, 0), HH - 1);
                s_off[t][j] = yc * WW + xc;
                s_wgt[t][j] = ok ? wxs[jx] * wys[jy] : 0.0f;
            }
        }
        const float invN = 1.0f / (float)MAXN;
        s_coef[t][0] = s * invN;
        s_coef[t][1] = s * (cx * cx + cy * cy) * invN;
        s_coef[t][2] = -2.0f * s * cx * invN;
        s_coef[t][3] = -2.0f * s * cy * invN;
    }
    __syncthreads();

    // One channel per thread: P[c][k] = sum_n coef[n][k] * cpf[c][n]
    const float* fc = feat + (size_t)t * HW;
    float a0 = 0.0f, a1 = 0.0f, a2 = 0.0f, a3 = 0.0f;
    for (int n = 0; n < MAXN; ++n) {
        float p = 0.0f;
        #pragma unroll
        for (int j = 0; j < 4; ++j)
            p += s_wgt[n][j] * fc[s_off[n][j]];
        a0 += s_coef[n][0] * p;
        a1 += s_coef[n][1] * p;
        a2 += s_coef[n][2] * p;
        a3 += s_coef[n][3] * p;
    }
    P[t * 4 + 0] = a0;
    P[t * 4 + 1] = a1;
    P[t * 4 + 2] = a2;
    P[t * 4 + 3] = a3;
}

// ---------------------------------------------------------------------------
// Stage 2: out[c,h,w] = P0*(w^2+h^2) + P1 + P2*w + P3*h as a K=4 WMMA GEMM.
// One V_WMMA_F32_16X16X4_F32 per 16x16 (c,w) tile. 4096 waves, 16 tiles each.
// ---------------------------------------------------------------------------
__global__ void __launch_bounds__(256)
comm_stage2(const float* __restrict__ P,   // [256][4]
            float* __restrict__ out)       // [256][256][256]
{
    const int gtid  = blockIdx.x * blockDim.x + threadIdx.x;
    const int wave  = gtid >> 5;              // 0..4095
    const int lane  = threadIdx.x & 31;
    const int ctile = wave & 15;              // 16 channel tiles
    const int h     = wave >> 4;              // 0..255
    const int half  = lane >> 4;              // 0|1 -> K pair {0,1} | {2,3}
    const int m     = lane & 15;

    // A fragment (16x4 f32): lane m holds rows M=m; VGPR0=K(2*half), VGPR1=K(2*half+1)
    const int c0 = ctile * 16 + m;
    v2f a;
    a.x = P[c0 * 4 + 2 * half + 0];
    a.y = P[c0 * 4 + 2 * half + 1];

    const float hf = (float)h;
    const float h2 = hf * hf;
    float* orow = out + (size_t)h * WW + (size_t)(ctile * 16 + 8 * half) * HW;

    for (int wt = 0; wt < 16; ++wt) {
        const float wf = (float)(wt * 16 + m);
        // B fragment (4x16 f32): lane m = column; halves hold K rows {0,1} / {2,3}
        // Q rows: Q0 = w^2 + h^2, Q1 = 1, Q2 = w, Q3 = h   (selects, no divergence)
        v2f b;
        b.x = (half == 0) ? (wf * wf + h2) : wf;
        b.y = (half == 0) ? 1.0f : hf;

        v8f c = {};
        v8f d = __builtin_amdgcn_wmma_f32_16x16x4_f32(
            /*neg_a=*/false, a, /*neg_b=*/false, b,
            /*c_mod=*/(short)0, c, /*reuse_a=*/false, /*reuse_b=*/false);

        // D layout: VGPR r -> row (r + 8*half), lane m -> column (wt*16 + m)
        float* op = orow + wt * 16 + m;
        #pragma unroll
        for (int r = 0; r < 8; ++r)
            op[(size_t)r * HW] = d[r];
    }
}

// ---------------------------------------------------------------------------
extern "C" void kernel_launch(void* const* d_in, const int* in_sizes, int n_in,
                              void* d_out, int out_size, void* d_ws, size_t ws_size,
                              hipStream_t stream)
{
    const float* box   = (const float*)d_in[0];  // [100,8,3]
    const float* score = (const float*)d_in[1];  // [100]
    const float* feat  = (const float*)d_in[2];  // [1,256,256,256]
    float*       out   = (float*)d_out;          // [1,256,256,256]
    float*       P     = (float*)d_ws;           // 256*4 floats scratch

    comm_stage1<<<1, 256, 0, stream>>>(box, score, feat, P);
    comm_stage2<<<512, 256, 0, stream>>>(P, out);
}